// GraphConv_1580547970207
// MI455X (gfx1250) — compile-verified
//
#include <hip/hip_runtime.h>

typedef __attribute__((ext_vector_type(2))) float    v2f;
typedef __attribute__((ext_vector_type(4))) float    v4f;
typedef __attribute__((ext_vector_type(8))) float    v8f;
typedef __attribute__((ext_vector_type(4))) unsigned u32x4;
typedef __attribute__((ext_vector_type(8))) unsigned u32x8;

#define BATCH  8
#define NNODE  8192
#define FDIM   128
#define TILE_M 128      // output rows per workgroup
#define TILE_K 16       // K chunk per pipeline stage
#define NCHUNK (NNODE / TILE_K)
#define LDA    20       // padded adj tile row (floats): TDM pads 16 dwords + 4 dwords
#define LDY    132      // leading dim for 128-wide y/z tile

#define AS_FLOATS (TILE_M * LDA)   // 2560 floats per adj buffer
#define XS_FLOATS (TILE_K * FDIM)  // 2048 floats per x buffer (linear copy)
#define XBASE     (3 * AS_FLOATS)  // float offset of first x buffer in Pool

// ---- CDNA5 Tensor Data Mover: one instruction DMAs a padded/strided tile ----
__device__ __forceinline__ void tdm_load(u32x4 g0, u32x8 g1) {
  asm volatile("tensor_load_to_lds %0, %1" :: "s"(g0), "s"(g1) : "memory");
}
__device__ __forceinline__ void wait_tensor2() {   // allow newest chunk (2 ops) in flight
  asm volatile("s_wait_tensorcnt 0x2" ::: "memory");
}
// ---- per-lane async DMA (used for the weight matrix) ----
__device__ __forceinline__ void async_b128(unsigned lds_byte_addr, const float* gaddr) {
  asm volatile("global_load_async_to_lds_b128 %0, %1, off"
               :: "v"(lds_byte_addr), "v"(gaddr) : "memory");
}
__device__ __forceinline__ void wait_async0() {
  asm volatile("s_wait_asynccnt 0x0" ::: "memory");
}

// D# group 0: count=1 | lds_addr | global byte addr (57b) | type=2 ("image")
__device__ __forceinline__ u32x4 tdm_g0(unsigned long long ga, unsigned lds) {
  u32x4 g;
  g.x = 1u;
  g.y = lds;
  g.z = (unsigned)ga;
  g.w = ((unsigned)(ga >> 32) & 0x01FFFFFFu) | (2u << 30);
  return g;
}

__launch_bounds__(256)
__global__ void graphconv_fused(const float* __restrict__ x,
                                const float* __restrict__ adj,
                                const float* __restrict__ weight,
                                const float* __restrict__ bias,
                                float* __restrict__ out) {
  // Pool overlay: GEMM1 = As0|As1|As2|Xs0|Xs1|Xs2 (13824 floats);
  //               GEMM2 = Ws[k][n] row-major (16384 floats)
  __shared__ float Pool[FDIM * FDIM];
  __shared__ float Ys[TILE_M * LDY];   // y tile, reused for z tile
  __shared__ float Sc[TILE_M];

  const int t     = threadIdx.x;
  const int lane  = t & 31;
  const int w     = t >> 5;      // wave 0..7
  const int wr    = w >> 1;      // 32-row group
  const int wc    = w & 1;       // 64-col group
  const int lmod  = lane & 15;
  const int lhalf = lane >> 4;   // selects K pair in fp32 WMMA fragment layout

  // wave id as a compiler-visible SGPR value -> scalar branch (s_cbranch_scc),
  // so waves 1..7 truly skip the TDM issue (TDM ignores EXEC masking!)
  const int wu = __builtin_amdgcn_readfirstlane(w);

  const int b  = blockIdx.x;             // batch fastest -> adj panel reuse in L2
  const int n0 = blockIdx.y * TILE_M;

  const unsigned poolBase = (unsigned)(size_t)(void*)Pool;   // LDS byte address

  // ---- D# group 1 descriptors (uniform constants) ----
  // adj tile: data_size=4B, pad_enable, pad_interval=16 dwords (code 3),
  // pad_amount=4 dwords (code 3); dims huge (no clip); tile 16 x 128; stride 8192
  const u32x8 g1_adj = { (2u << 16) | (1u << 20) | (3u << 22) | (3u << 25),
                         0u, 0x4000u, 0x4000u | (16u << 16),
                         128u, 8192u, 0u, 0u };
  // x chunk: data_size=4B, no pad, 1D tile of 2048 elements
  const u32x8 g1_x   = { (2u << 16),
                         0u, 0x4000u, 0x4000u | (2048u << 16),
                         0u, 2048u, 0u, 0u };

  const unsigned long long gadj =
      (unsigned long long)(size_t)adj + (unsigned long long)n0 * NNODE * 4ull;
  const unsigned long long gxb =
      (unsigned long long)(size_t)x + (unsigned long long)b * NNODE * FDIM * 4ull;

  // issue chunk c into buffers given by float offsets offA (adj) / offX (x)
  auto issue_chunk = [&](int c, unsigned offA, unsigned offX) {
    tdm_load(tdm_g0(gadj + (unsigned long long)c * (TILE_K * 4),
                    poolBase + offA * 4u), g1_adj);
    tdm_load(tdm_g0(gxb + (unsigned long long)c * (TILE_K * FDIM * 4),
                    poolBase + offX * 4u), g1_x);
  };

  v8f acc[2][4];
#pragma unroll
  for (int sr = 0; sr < 2; ++sr)
#pragma unroll
    for (int sc = 0; sc < 4; ++sc) acc[sr][sc] = (v8f)0.0f;

  auto gemm1_step = [&](const float* As, const float* Xs) {
#pragma unroll
    for (int kk = 0; kk < TILE_K; kk += 4) {
      v2f av[2], bv[4];
#pragma unroll
      for (int sr = 0; sr < 2; ++sr) {
        int row = 32 * wr + 16 * sr + lmod;
        av[sr] = *(const v2f*)(As + row * LDA + kk + 2 * lhalf);
      }
#pragma unroll
      for (int sc = 0; sc < 4; ++sc) {
        int col = 64 * wc + 16 * sc + lmod;
        int kb  = (kk + 2 * lhalf) * FDIM + col;     // ds_load_2addr pair
        v2f bb; bb.x = Xs[kb]; bb.y = Xs[kb + FDIM];
        bv[sc] = bb;
      }
#pragma unroll
      for (int sr = 0; sr < 2; ++sr)
#pragma unroll
        for (int sc = 0; sc < 4; ++sc)
          acc[sr][sc] = __builtin_amdgcn_wmma_f32_16x16x4_f32(
              false, av[sr], false, bv[sc], (short)0, acc[sr][sc], false, false);
    }
  };

  // ================= GEMM 1: y = adj @ x[b], TDM 3-stage pipeline =================
  if (wu == 0) {
    issue_chunk(0, 0u,        XBASE);
    issue_chunk(1, AS_FLOATS, XBASE + XS_FLOATS);
    wait_tensor2();          // chunk 0 resident; chunk 1 may be in flight
  }
  __syncthreads();

  // rotating integer float-offsets (no pointer tables -> no LDS addr constants)
  unsigned offA_cur = 0,             offX_cur = XBASE;
  unsigned offA_nxt = 2 * AS_FLOATS, offX_nxt = XBASE + 2 * XS_FLOATS;

  for (int i = 0; i < NCHUNK; ++i) {
    if (wu == 0 && i + 2 < NCHUNK) issue_chunk(i + 2, offA_nxt, offX_nxt);
    gemm1_step(Pool + offA_cur, Pool + offX_cur);
    if (wu == 0) wait_tensor2();   // chunk i+1 resident before next iteration
    __syncthreads();
    offA_cur = (offA_cur == 2 * AS_FLOATS) ? 0u : offA_cur + AS_FLOATS;
    offA_nxt = (offA_nxt == 2 * AS_FLOATS) ? 0u : offA_nxt + AS_FLOATS;
    offX_cur = (offX_cur == XBASE + 2 * XS_FLOATS) ? XBASE : offX_cur + XS_FLOATS;
    offX_nxt = (offX_nxt == XBASE + 2 * XS_FLOATS) ? XBASE : offX_nxt + XS_FLOATS;
  }

  // ---- kick Ws DMA into Pool (row-major [k][n]); overlaps the epilogue below
#pragma unroll
  for (int j = 0; j < 16; ++j) {
    int v = t + 256 * j;                 // float4 index over 128*128/4
    async_b128(poolBase + (unsigned)v * 16u, weight + v * 4);
  }

  // ---- spill y accumulators to LDS (for re-fragmentation as A)
#pragma unroll
  for (int sr = 0; sr < 2; ++sr)
#pragma unroll
    for (int sc = 0; sc < 4; ++sc) {
      int col = 64 * wc + 16 * sc + lmod;
#pragma unroll
      for (int i = 0; i < 8; ++i) {
        int row = 32 * wr + 16 * sr + i + 8 * lhalf;
        Ys[row * LDY + col] = acc[sr][sc][i];
      }
    }
  __syncthreads();

  // ---- self add: Ys += x tile (coalesced float4), overlapped with Ws DMA
#pragma unroll
  for (int j = 0; j < 16; ++j) {
    int v  = t + 256 * j;
    int r  = v >> 5;
    int fq = v & 31;
    v4f xv = *(const v4f*)(x + ((long)b * NNODE + n0 + r) * FDIM + fq * 4);
    v4f* yp = (v4f*)(Ys + r * LDY + fq * 4);
    v4f yv = *yp;
    yv.x += xv.x; yv.y += xv.y; yv.z += xv.z; yv.w += xv.w;
    *yp = yv;
  }
  wait_async0();       // Ws resident
  __syncthreads();

  // ================= GEMM 2: z = y @ weight =================
  v8f z[2][4];
#pragma unroll
  for (int sr = 0; sr < 2; ++sr)
#pragma unroll
    for (int sc = 0; sc < 4; ++sc) z[sr][sc] = (v8f)0.0f;

#pragma unroll 4
  for (int kk = 0; kk < FDIM; kk += 4) {
    v2f av[2], bv[4];
#pragma unroll
    for (int sr = 0; sr < 2; ++sr) {
      int row = 32 * wr + 16 * sr + lmod;
      av[sr] = *(const v2f*)(Ys + row * LDY + kk + 2 * lhalf);
    }
#pragma unroll
    for (int sc = 0; sc < 4; ++sc) {
      int col = 64 * wc + 16 * sc + lmod;
      int kb  = (kk + 2 * lhalf) * FDIM + col;
      v2f bb; bb.x = Pool[kb]; bb.y = Pool[kb + FDIM];
      bv[sc] = bb;
    }
#pragma unroll
    for (int sr = 0; sr < 2; ++sr)
#pragma unroll
      for (int sc = 0; sc < 4; ++sc)
        z[sr][sc] = __builtin_amdgcn_wmma_f32_16x16x4_f32(
            false, av[sr], false, bv[sc], (short)0, z[sr][sc], false, false);
  }

  __syncthreads();   // all Ys reads done before overwrite with z

  // ---- bias add, store z tile to LDS
#pragma unroll
  for (int sc = 0; sc < 4; ++sc) {
    int col  = 64 * wc + 16 * sc + lmod;
    float bb = bias[col];
#pragma unroll
    for (int sr = 0; sr < 2; ++sr)
#pragma unroll
      for (int i = 0; i < 8; ++i) {
        int row = 32 * wr + 16 * sr + i + 8 * lhalf;
        Ys[row * LDY + col] = z[sr][sc][i] + bb;
      }
  }
  __syncthreads();

  // ---- per-row L2 norm -> scale
  if (t < TILE_M) {
    float s = 0.0f;
#pragma unroll 8
    for (int fq = 0; fq < 32; ++fq) {
      v4f yv = *(const v4f*)(Ys + t * LDY + fq * 4);
      s += yv.x * yv.x + yv.y * yv.y + yv.z * yv.z + yv.w * yv.w;
    }
    Sc[t] = 1.0f / fmaxf(sqrtf(s), 1e-12f);
  }
  __syncthreads();

  // ---- scaled, coalesced float4 stores
#pragma unroll
  for (int j = 0; j < 16; ++j) {
    int v  = t + 256 * j;
    int r  = v >> 5;
    int fq = v & 31;
    v4f o = *(const v4f*)(Ys + r * LDY + fq * 4);
    float s = Sc[r];
    o.x *= s; o.y *= s; o.z *= s; o.w *= s;
    *(v4f*)(out + ((long)b * NNODE + n0 + r) * FDIM + fq * 4) = o;
  }
}

extern "C" void kernel_launch(void* const* d_in, const int* in_sizes, int n_in,
                              void* d_out, int out_size, void* d_ws, size_t ws_size,
                              hipStream_t stream) {
  const float* x      = (const float*)d_in[0];
  const float* adj    = (const float*)d_in[1];
  const float* weight = (const float*)d_in[2];
  const float* bias   = (const float*)d_in[3];
  float* out          = (float*)d_out;

  dim3 grid(BATCH, NNODE / TILE_M);   // batch fastest -> adj panel shared in L2
  dim3 block(256);
  graphconv_fused<<<grid, block, 0, stream>>>(x, adj, weight, bias, out);
}